// LearnedEarlyExitStudent_75857712382066
// MI455X (gfx1250) — compile-verified
//
#include <hip/hip_runtime.h>
#include <hip/hip_bf16.h>
#include <math.h>

// ---------------------------------------------------------------------------
// LearnedEarlyExitStudent: packet-embed -> 3x Mamba -> heads, all fp32.
// GEMMs use V_WMMA_F32_16X16X4_F32 (full fp32 precision; problem is
// bandwidth/latency bound at ~43 GFLOP / ~150MB, so low precision buys nothing).
// ---------------------------------------------------------------------------

typedef float v2f __attribute__((ext_vector_type(2)));
typedef float v8f __attribute__((ext_vector_type(8)));

#define NTOK   16384      // B*L = 512*32
#define DM     256
#define DI     512
#define DS     16
#define DBC_W  48

// ---------------- generic f32 WMMA GEMM: C[M,N] = A[M,K]@B[K,N] (+bias,act) --
// act: 0 = none, 1 = softplus
__global__ __launch_bounds__(256)
void gemm_wmma_f32(const float* __restrict__ A, int lda,
                   const float* __restrict__ Bm, int ldb,
                   float* __restrict__ Cout, int ldc,
                   const float* __restrict__ bias,
                   int M, int N, int K, int act)
{
    int wave = (int)((blockIdx.x * blockDim.x + threadIdx.x) >> 5);
    int lane = threadIdx.x & 31;
    int tilesN = N >> 4;
    int tilesM = M >> 4;
    if (wave >= tilesM * tilesN) return;     // wave-uniform: EXEC stays all-1s
    int tm = wave / tilesN, tn = wave % tilesN;
    int m0 = tm << 4, n0 = tn << 4;

    int half = lane >> 4;      // 0: K pair {0,1}; 1: K pair {2,3}
    int l15  = lane & 15;

    const float* Arow = A  + (size_t)(m0 + l15) * lda + (half ? 2 : 0);
    const float* Bcol = Bm + (size_t)(half ? 2 : 0) * ldb + n0 + l15;

    v8f acc = {};
    for (int k0 = 0; k0 < K; k0 += 4) {
        v2f a, b;
        a.x = Arow[k0 + 0];
        a.y = Arow[k0 + 1];
        b.x = Bcol[(size_t)(k0 + 0) * ldb];
        b.y = Bcol[(size_t)(k0 + 1) * ldb];
        acc = __builtin_amdgcn_wmma_f32_16x16x4_f32(
                  /*neg_a=*/false, a, /*neg_b=*/false, b,
                  /*c_mod=*/(short)0, acc, /*reuse_a=*/false, /*reuse_b=*/false);
    }

    float bv = bias ? bias[n0 + l15] : 0.0f;
    #pragma unroll
    for (int v = 0; v < 8; ++v) {
        int row = v + (half ? 8 : 0);
        float r = acc[v] + bv;
        if (act == 1)   // softplus, numerically stable
            r = (r > 0.0f) ? (r + log1pf(__expf(-r))) : log1pf(__expf(r));
        Cout[(size_t)(m0 + row) * ldc + n0 + l15] = r;
    }
}

// ---------------- embedding concat: cat[tok,136] ----------------------------
__global__ __launch_bounds__(256)
void embed_cat(const float* __restrict__ x,
               const float* __restrict__ emb_proto, const float* __restrict__ emb_flags,
               const float* __restrict__ emb_dir,
               const float* __restrict__ plw, const float* __restrict__ plb,
               const float* __restrict__ piw, const float* __restrict__ pib,
               float* __restrict__ cat)
{
    int idx = blockIdx.x * blockDim.x + threadIdx.x;
    if (idx >= NTOK * 136) return;
    int tok = idx / 136;
    int c   = idx - tok * 136;
    const float* xt = x + (size_t)tok * 5;
    float v;
    if (c < 32) {
        int p = (int)xt[0]; p = p < 0 ? 0 : (p > 255 ? 255 : p);
        v = emb_proto[p * 32 + c];
    } else if (c < 64) {
        int f = (int)xt[2]; f = f < 0 ? 0 : (f > 63 ? 63 : f);
        v = emb_flags[f * 32 + (c - 32)];
    } else if (c < 72) {
        int d = (int)xt[4]; d = d < 0 ? 0 : (d > 1 ? 1 : d);
        v = emb_dir[d * 8 + (c - 64)];
    } else if (c < 104) {
        v = xt[1] * plw[c - 72] + plb[c - 72];
    } else {
        v = xt[3] * piw[c - 104] + pib[c - 104];
    }
    cat[idx] = v;
}

// ---------------- in-place LayerNorm over 256-wide rows (wave per row) ------
__global__ __launch_bounds__(256)
void layernorm_rows(float* __restrict__ data,
                    const float* __restrict__ g, const float* __restrict__ b,
                    int rows)
{
    int wave = (int)((blockIdx.x * blockDim.x + threadIdx.x) >> 5);
    int lane = threadIdx.x & 31;
    if (wave >= rows) return;
    float* row = data + (size_t)wave * DM;
    float vals[8];
    float s = 0.0f;
    #pragma unroll
    for (int i = 0; i < 8; ++i) { vals[i] = row[lane * 8 + i]; s += vals[i]; }
    #pragma unroll
    for (int o = 16; o > 0; o >>= 1) s += __shfl_xor(s, o, 32);
    float mean = s * (1.0f / DM);
    float sv = 0.0f;
    #pragma unroll
    for (int i = 0; i < 8; ++i) { float d = vals[i] - mean; sv += d * d; }
    #pragma unroll
    for (int o = 16; o > 0; o >>= 1) sv += __shfl_xor(sv, o, 32);
    float inv = rsqrtf(sv * (1.0f / DM) + 1e-5f);
    #pragma unroll
    for (int i = 0; i < 8; ++i) {
        int c = lane * 8 + i;
        row[c] = (vals[i] - mean) * inv * g[c] + b[c];
    }
}

// ---------------- causal depthwise conv (k=4) + SiLU ------------------------
// xi = xz[:, 0:512] of xz[tok,1024]
__global__ __launch_bounds__(256)
void conv_silu(const float* __restrict__ xz, const float* __restrict__ w,
               const float* __restrict__ bias, float* __restrict__ xc)
{
    int idx = blockIdx.x * blockDim.x + threadIdx.x;
    if (idx >= NTOK * DI) return;
    int tok = idx >> 9;
    int d   = idx & (DI - 1);
    int b   = tok >> 5;
    int l   = tok & 31;
    float acc = bias[d];
    #pragma unroll
    for (int j = 0; j < 4; ++j) {
        int ls = l - 3 + j;
        if (ls >= 0)
            acc += w[d * 4 + j] * xz[(size_t)((b << 5) + ls) * 1024 + d];
    }
    float sg = 1.0f / (1.0f + __expf(-acc));
    xc[idx] = acc * sg;
}

// ---------------- selective scan: thread per (b,d), h[16] in regs -----------
// dt_y holds softplus(dt) on entry; overwritten with y*silu(z) (safe: each
// thread reads dt[b,t,d] before writing y[b,t,d]).
__global__ __launch_bounds__(256)
void scan_kernel(const float* __restrict__ xz, const float* __restrict__ xc,
                 const float* __restrict__ dbc, float* __restrict__ dt_y,
                 const float* __restrict__ A_log, const float* __restrict__ Dp)
{
    int idx = blockIdx.x * blockDim.x + threadIdx.x;
    if (idx >= 512 * DI) return;
    int b = idx >> 9;
    int d = idx & (DI - 1);

    float a[DS], h[DS];
    #pragma unroll
    for (int s = 0; s < DS; ++s) { a[s] = -__expf(A_log[d * DS + s]); h[s] = 0.0f; }
    float Dd = Dp[d];

    for (int t = 0; t < 32; ++t) {
        int tok = (b << 5) + t;
        float dtv = dt_y[(size_t)tok * DI + d];
        float xcv = xc  [(size_t)tok * DI + d];
        const float* bc = dbc + (size_t)tok * DBC_W;
        float y = 0.0f;
        #pragma unroll
        for (int s = 0; s < DS; ++s) {
            float dA = __expf(dtv * a[s]);
            h[s] = dA * h[s] + dtv * bc[16 + s] * xcv;
            y += h[s] * bc[32 + s];
        }
        y += xcv * Dd;
        float z  = xz[(size_t)tok * 1024 + DI + d];
        float sz = z / (1.0f + __expf(-z));
        dt_y[(size_t)tok * DI + d] = y * sz;
    }
}

// ---------------- heads -----------------------------------------------------
__global__ __launch_bounds__(256)
void halt_kernel(const float* __restrict__ feats, const float* __restrict__ hw,
                 const float* __restrict__ hb, float* __restrict__ out)
{
    int wave = (int)((blockIdx.x * blockDim.x + threadIdx.x) >> 5);
    int lane = threadIdx.x & 31;
    if (wave >= NTOK) return;
    const float* row = feats + (size_t)wave * DM;
    float s = 0.0f;
    #pragma unroll
    for (int i = 0; i < 8; ++i) { int c = lane * 8 + i; s += row[c] * hw[c]; }
    #pragma unroll
    for (int o = 16; o > 0; o >>= 1) s += __shfl_xor(s, o, 32);
    if (lane == 0) out[wave] = 1.0f / (1.0f + __expf(-(s + hb[0])));
}

// LN(row) then 2-class logits; used for both final and early heads.
__device__ __forceinline__
void ln_logits(const float* row, int lane,
               const float* g, const float* bt,
               const float* w2 /*[256,2]*/, const float* b2 /*[2]*/,
               float* out2)
{
    float vals[8];
    float s = 0.0f;
    #pragma unroll
    for (int i = 0; i < 8; ++i) { vals[i] = row[lane * 8 + i]; s += vals[i]; }
    #pragma unroll
    for (int o = 16; o > 0; o >>= 1) s += __shfl_xor(s, o, 32);
    float mean = s * (1.0f / DM);
    float sv = 0.0f;
    #pragma unroll
    for (int i = 0; i < 8; ++i) { float d = vals[i] - mean; sv += d * d; }
    #pragma unroll
    for (int o = 16; o > 0; o >>= 1) sv += __shfl_xor(sv, o, 32);
    float inv = rsqrtf(sv * (1.0f / DM) + 1e-5f);
    float l0 = 0.0f, l1 = 0.0f;
    #pragma unroll
    for (int i = 0; i < 8; ++i) {
        int c = lane * 8 + i;
        float xn = (vals[i] - mean) * inv * g[c] + bt[c];
        l0 += xn * w2[c * 2 + 0];
        l1 += xn * w2[c * 2 + 1];
    }
    #pragma unroll
    for (int o = 16; o > 0; o >>= 1) { l0 += __shfl_xor(l0, o, 32); l1 += __shfl_xor(l1, o, 32); }
    if (lane == 0) { out2[0] = l0 + b2[0]; out2[1] = l1 + b2[1]; }
}

__global__ __launch_bounds__(256)
void final_kernel(const float* __restrict__ feats,
                  const float* __restrict__ g, const float* __restrict__ bt,
                  const float* __restrict__ fw, const float* __restrict__ fb,
                  float* __restrict__ out)
{
    int wave = (int)((blockIdx.x * blockDim.x + threadIdx.x) >> 5);
    int lane = threadIdx.x & 31;
    if (wave >= 512) return;
    const float* row = feats + (size_t)(wave * 32 + 31) * DM;
    ln_logits(row, lane, g, bt, fw, fb, out + wave * 2);
}

__global__ __launch_bounds__(256)
void early_kernel(const float* __restrict__ feats,
                  const float* __restrict__ g, const float* __restrict__ bt,
                  const float* __restrict__ ew /*[18,256,2]*/,
                  const float* __restrict__ eb /*[18,2]*/,
                  float* __restrict__ out /*[18,512,2]*/)
{
    int wave = (int)((blockIdx.x * blockDim.x + threadIdx.x) >> 5);
    int lane = threadIdx.x & 31;
    if (wave >= 18 * 512) return;
    int p = wave / 512;
    int b = wave - p * 512;
    const float* row = feats + (size_t)(b * 32 + 3 + p) * DM;   // MIN_EXIT = 3
    ln_logits(row, lane, g, bt, ew + (size_t)p * DM * 2, eb + p * 2,
              out + (size_t)p * 1024 + b * 2);
}

// ---------------------------------------------------------------------------
extern "C" void kernel_launch(void* const* d_in, const int* in_sizes, int n_in,
                              void* d_out, int out_size, void* d_ws, size_t ws_size,
                              hipStream_t stream)
{
    (void)in_sizes; (void)n_in; (void)out_size; (void)ws_size;

    const float* x         = (const float*)d_in[0];
    const float* emb_proto = (const float*)d_in[1];
    const float* emb_flags = (const float*)d_in[2];
    const float* emb_dir   = (const float*)d_in[3];
    const float* plw       = (const float*)d_in[4];
    const float* plb       = (const float*)d_in[5];
    const float* piw       = (const float*)d_in[6];
    const float* pib       = (const float*)d_in[7];
    const float* fusion_w  = (const float*)d_in[8];
    const float* fusion_b  = (const float*)d_in[9];
    const float* emb_ln_g  = (const float*)d_in[10];
    const float* emb_ln_b  = (const float*)d_in[11];
    // layers: 9 tensors each starting at index 12
    const float* norm_g    = (const float*)d_in[39];
    const float* norm_b    = (const float*)d_in[40];
    const float* early_w   = (const float*)d_in[41];
    const float* early_b   = (const float*)d_in[42];
    const float* final_w   = (const float*)d_in[43];
    const float* final_b   = (const float*)d_in[44];
    const float* halt_w    = (const float*)d_in[45];
    const float* halt_b    = (const float*)d_in[46];

    float* out = (float*)d_out;          // [final 1024 | halt 16384 | early 18432]

    // workspace layout (floats); y aliases dt, cat aliases xz
    float* ws    = (float*)d_ws;
    float* feats = ws;                               // 16384*256
    float* xz    = feats + (size_t)NTOK * DM;        // 16384*1024
    float* xc    = xz    + (size_t)NTOK * 1024;      // 16384*512
    float* dbc   = xc    + (size_t)NTOK * DI;        // 16384*48
    float* dt    = dbc   + (size_t)NTOK * DBC_W;     // 16384*512
    float* cat   = xz;                               // alias

    const int BLK = 256;
    auto gemm_blocks = [](int M, int N) { return ((M >> 4) * (N >> 4) + 7) / 8; };

    // ---- embedding ----
    embed_cat<<<(NTOK * 136 + BLK - 1) / BLK, BLK, 0, stream>>>(
        x, emb_proto, emb_flags, emb_dir, plw, plb, piw, pib, cat);
    gemm_wmma_f32<<<gemm_blocks(NTOK, DM), BLK, 0, stream>>>(
        cat, 136, fusion_w, DM, feats, DM, fusion_b, NTOK, DM, 136, 0);
    layernorm_rows<<<(NTOK + 7) / 8, BLK, 0, stream>>>(feats, emb_ln_g, emb_ln_b, NTOK);

    // ---- mamba layers ----
    for (int i = 0; i < 3; ++i) {
        const float* in_proj_w  = (const float*)d_in[12 + 9 * i + 0];
        const float* conv_w     = (const float*)d_in[12 + 9 * i + 1];
        const float* conv_b     = (const float*)d_in[12 + 9 * i + 2];
        const float* x_proj_w   = (const float*)d_in[12 + 9 * i + 3];
        const float* dt_proj_w  = (const float*)d_in[12 + 9 * i + 4];
        const float* dt_proj_b  = (const float*)d_in[12 + 9 * i + 5];
        const float* A_log      = (const float*)d_in[12 + 9 * i + 6];
        const float* Dp         = (const float*)d_in[12 + 9 * i + 7];
        const float* out_proj_w = (const float*)d_in[12 + 9 * i + 8];

        // xz = feats @ in_proj_w   [16384,256]@[256,1024]
        gemm_wmma_f32<<<gemm_blocks(NTOK, 1024), BLK, 0, stream>>>(
            feats, DM, in_proj_w, 1024, xz, 1024, nullptr, NTOK, 1024, DM, 0);
        // xc = silu(causal depthwise conv(xi))
        conv_silu<<<(NTOK * DI + BLK - 1) / BLK, BLK, 0, stream>>>(xz, conv_w, conv_b, xc);
        // dbc = xc @ x_proj_w      [16384,512]@[512,48]
        gemm_wmma_f32<<<gemm_blocks(NTOK, DBC_W), BLK, 0, stream>>>(
            xc, DI, x_proj_w, DBC_W, dbc, DBC_W, nullptr, NTOK, DBC_W, DI, 0);
        // dt = softplus(dbc[:, :16] @ dt_proj_w + b)   [16384,16]@[16,512]
        gemm_wmma_f32<<<gemm_blocks(NTOK, DI), BLK, 0, stream>>>(
            dbc, DBC_W, dt_proj_w, DI, dt, DI, dt_proj_b, NTOK, DI, DS, 1);
        // selective scan; writes y*silu(z) in place of dt
        scan_kernel<<<(512 * DI + BLK - 1) / BLK, BLK, 0, stream>>>(
            xz, xc, dbc, dt, A_log, Dp);
        // feats = y @ out_proj_w   [16384,512]@[512,256]
        gemm_wmma_f32<<<gemm_blocks(NTOK, DM), BLK, 0, stream>>>(
            dt, DI, out_proj_w, DM, feats, DM, nullptr, NTOK, DM, DI, 0);
    }

    // ---- heads ----
    final_kernel<<<(512 + 7) / 8, BLK, 0, stream>>>(
        feats, norm_g, norm_b, final_w, final_b, out);                    // [0,1024)
    halt_kernel<<<(NTOK + 7) / 8, BLK, 0, stream>>>(
        feats, halt_w, halt_b, out + 1024);                               // [1024,17408)
    early_kernel<<<(18 * 512 + 7) / 8, BLK, 0, stream>>>(
        feats, norm_g, norm_b, early_w, early_b, out + 1024 + NTOK);      // [17408,35840)
}